// LowerAdjTransformer_11596411699254
// MI455X (gfx1250) — compile-verified
//
#include <hip/hip_runtime.h>
#include <hip/hip_bf16.h>
#include <math.h>

// Problem constants (fixed by the reference)
constexpr int NN   = 50000;
constexpr int EE   = 800000;
constexpr int DIN  = 64;
constexpr int DOUT = 32;
constexpr int GEO  = 16;
constexpr int HH   = 3;
constexpr int HD   = 96;   // H*DOUT
constexpr int EFK  = 145;  // DIN*2 + GEO + 1
constexpr int EFKP = 148;  // padded to multiple of 4

typedef __attribute__((ext_vector_type(2))) float v2f;
typedef __attribute__((ext_vector_type(8))) float v8f;

__device__ __forceinline__ v8f wmma_f32(v2f a, v2f b, v8f c) {
  // D = A(16x4,f32) x B(4x16,f32) + C(16x16,f32)
  return __builtin_amdgcn_wmma_f32_16x16x4_f32(false, a, false, b, (short)0, c,
                                               false, false);
}

// ---------------------------------------------------------------- prep edges
__global__ void k_prep(const float* __restrict__ adj, int* __restrict__ src,
                       int* __restrict__ dst, float* __restrict__ dw) {
  int e = blockIdx.x * blockDim.x + threadIdx.x;
  if (e >= EE) return;
  src[e] = (int)adj[e * 4 + 0];
  dst[e] = (int)adj[e * 4 + 1];
  dw[e]  = adj[e * 4 + 3];
}

// Zero-padded copy of edge_W: (145x32) -> (148x32), rows 145..147 = 0
__global__ void k_padW(const float* __restrict__ W, float* __restrict__ Wp) {
  int t = blockIdx.x * blockDim.x + threadIdx.x;
  if (t >= EFKP * DOUT) return;
  int r = t / DOUT;
  Wp[t] = (r < EFK) ? W[t] : 0.f;
}

// CSR offsets: dst is sorted ascending; lower_bound per node
__global__ void k_rowstart(const int* __restrict__ dst, int* __restrict__ rs) {
  int n = blockIdx.x * blockDim.x + threadIdx.x;
  if (n > NN) return;
  int lo = 0, hi = EE;
  while (lo < hi) {
    int mid = (lo + hi) >> 1;
    if (dst[mid] < n) lo = mid + 1; else hi = mid;
  }
  rs[n] = lo;
}

// ------------------------------------------------- edge MLP: e_attr = ef @ W + b
__global__ __launch_bounds__(32)
void k_edge_attr(const float* __restrict__ h, const float* __restrict__ geo,
                 const int* __restrict__ src, const int* __restrict__ dst,
                 const float* __restrict__ dw, const float* __restrict__ Wp,
                 const float* __restrict__ bias, float* __restrict__ eattr) {
  __shared__ float As[16][EFKP];
  const int lane = threadIdx.x;
  const int e0 = blockIdx.x * 16;
  for (int idx = lane; idx < 16 * EFKP; idx += 32) {
    int r = idx / EFKP, f = idx % EFKP;
    int e = e0 + r;
    float val = 0.f;
    if (e < EE) {
      if (f < DIN)            val = h[dst[e] * DIN + f];
      else if (f < 2 * DIN)   val = h[src[e] * DIN + (f - DIN)] * dw[e];
      else if (f == 2 * DIN)  val = dw[e];
      else if (f < EFK)       val = geo[e * GEO + (f - 2 * DIN - 1)];
    }
    As[r][f] = val;
  }
  __syncthreads();
  const int lh = lane >> 4, lm = lane & 15;
  v8f c0 = {}; v8f c1 = {};
  for (int k0 = 0; k0 < EFKP; k0 += 4) {
    int kr = k0 + 2 * lh;
    v2f a; a.x = As[lm][kr]; a.y = As[lm][kr + 1];
    v2f b0, b1;
    b0.x = Wp[kr * DOUT + lm];
    b0.y = Wp[(kr + 1) * DOUT + lm];
    b1.x = Wp[kr * DOUT + 16 + lm];
    b1.y = Wp[(kr + 1) * DOUT + 16 + lm];
    c0 = wmma_f32(a, b0, c0);
    c1 = wmma_f32(a, b1, c1);
  }
  for (int r = 0; r < 8; ++r) {
    int e = e0 + 8 * lh + r;
    if (e < EE) {
      eattr[e * DOUT + lm]      = c0[r] + bias[lm];
      eattr[e * DOUT + 16 + lm] = c1[r] + bias[16 + lm];
    }
  }
}

// --------------------------------------- generic dense node GEMM: Y = X@W + b
// One A-tile staging per block; all Nout/16 column tiles computed internally.
__global__ __launch_bounds__(32)
void k_gemm_node(const float* __restrict__ X, const float* __restrict__ W,
                 const float* __restrict__ bias, float* __restrict__ Y,
                 int n, int K, int Nout) {
  extern __shared__ float As[]; // 16*K
  const int lane = threadIdx.x;
  const int r0 = blockIdx.x * 16;
  for (int idx = lane; idx < 16 * K; idx += 32) {
    int r = idx / K, f = idx % K;
    int row = r0 + r;
    As[r * K + f] = (row < n) ? X[row * K + f] : 0.f;
  }
  __syncthreads();
  const int lh = lane >> 4, lm = lane & 15;
  for (int ct = 0; ct < Nout / 16; ++ct) {
    v8f c = {};
    for (int k0 = 0; k0 < K; k0 += 4) {
      int kr = k0 + 2 * lh;
      v2f a; a.x = As[lm * K + kr]; a.y = As[lm * K + kr + 1];
      v2f b; b.x = W[kr * Nout + ct * 16 + lm];
      b.y = W[(kr + 1) * Nout + ct * 16 + lm];
      c = wmma_f32(a, b, c);
    }
    for (int r = 0; r < 8; ++r) {
      int row = r0 + 8 * lh + r;
      if (row < n) Y[row * Nout + ct * 16 + lm] = c[r] + bias[ct * 16 + lm];
    }
  }
}

// ---------------- attention scores: eh = e_attr@We (WMMA), alpha = q[dst].(k[src]+eh)
__global__ __launch_bounds__(32)
void k_alpha(const float* __restrict__ eattr, const float* __restrict__ We,
             const float* __restrict__ q, const float* __restrict__ kk,
             const int* __restrict__ src, const int* __restrict__ dst,
             float* __restrict__ alpha) {
  __shared__ float As[16][DOUT];
  __shared__ float Eh[16][HD];
  const int lane = threadIdx.x;
  const int e0 = blockIdx.x * 16;
  for (int idx = lane; idx < 16 * DOUT; idx += 32) {
    int r = idx >> 5, f = idx & 31;
    int e = e0 + r;
    As[r][f] = (e < EE) ? eattr[e * DOUT + f] : 0.f;
  }
  __syncthreads();
  const int lh = lane >> 4, lm = lane & 15;
  for (int ct = 0; ct < 6; ++ct) {
    v8f c = {};
    for (int k0 = 0; k0 < DOUT; k0 += 4) {
      int kr = k0 + 2 * lh;
      v2f a; a.x = As[lm][kr]; a.y = As[lm][kr + 1];
      v2f b; b.x = We[kr * HD + ct * 16 + lm];
      b.y = We[(kr + 1) * HD + ct * 16 + lm];
      c = wmma_f32(a, b, c);
    }
    for (int r = 0; r < 8; ++r) Eh[8 * lh + r][ct * 16 + lm] = c[r];
  }
  __syncthreads();
  for (int t = lane; t < 16 * HH; t += 32) {
    int el = t / HH, hh = t % HH;
    int e = e0 + el;
    if (e < EE) {
      int s = src[e], d = dst[e];
      const float* qp = q + (size_t)d * HD + hh * DOUT;
      const float* kp = kk + (size_t)s * HD + hh * DOUT;
      const float* ep = &Eh[el][hh * DOUT];
      float acc = 0.f;
      for (int j = 0; j < DOUT; ++j) acc += qp[j] * (kp[j] + ep[j]);
      alpha[(size_t)e * HH + hh] = acc * 0.17677669529663687f; // 1/sqrt(32)
    }
  }
}

// ------------------------------------------------ segment max + sum(exp)
__global__ void k_stats(const float* __restrict__ alpha, const int* __restrict__ rs,
                        float* __restrict__ amax, float* __restrict__ den) {
  int t = blockIdx.x * blockDim.x + threadIdx.x;
  if (t >= NN * HH) return;
  int n = t / HH, hh = t % HH;
  int b = rs[n], e = rs[n + 1];
  float m = -3.4e38f;
  for (int i = b; i < e; ++i) m = fmaxf(m, alpha[(size_t)i * HH + hh]);
  float s = 0.f;
  for (int i = b; i < e; ++i) s += expf(alpha[(size_t)i * HH + hh] - m);
  amax[t] = m;
  den[t] = s;
}

// ------------------------------------------------ alpha -> normalized weight a
__global__ void k_norm(float* __restrict__ alpha, const int* __restrict__ dst,
                       const float* __restrict__ amax, const float* __restrict__ den) {
  int t = blockIdx.x * blockDim.x + threadIdx.x;
  if (t >= EE * HH) return;
  int e = t / HH, hh = t % HH;
  int d = dst[e];
  float dn = den[d * HH + hh];
  alpha[t] = (dn > 0.f) ? expf(alpha[t] - amax[d * HH + hh]) / dn : 0.f;
}

// --------- aggregate: meanV = mean_h sum_e a*v[src]; accW = per-head sum_e a*e_attr
__global__ void k_aggregate(const float* __restrict__ alpha, const int* __restrict__ rs,
                            const int* __restrict__ src, const float* __restrict__ v,
                            const float* __restrict__ eattr,
                            float* __restrict__ meanV, float* __restrict__ accW) {
  int t = blockIdx.x * blockDim.x + threadIdx.x;
  if (t >= NN * DOUT) return;
  int n = t >> 5, d = t & 31;
  int b = rs[n], e = rs[n + 1];
  float av = 0.f, w0 = 0.f, w1 = 0.f, w2 = 0.f;
  for (int i = b; i < e; ++i) {
    if (i + 1 < e) {
      // prefetch next gathered v row (random access over src)
      __builtin_prefetch(v + (size_t)src[i + 1] * HD + d, 0, 1);
    }
    float a0 = alpha[(size_t)i * HH];
    float a1 = alpha[(size_t)i * HH + 1];
    float a2 = alpha[(size_t)i * HH + 2];
    int s = src[i];
    float ea = eattr[(size_t)i * DOUT + d];
    w0 += a0 * ea; w1 += a1 * ea; w2 += a2 * ea;
    const float* vp = v + (size_t)s * HD;
    av += a0 * vp[d] + a1 * vp[32 + d] + a2 * vp[64 + d];
  }
  meanV[t] = av * (1.f / 3.f);
  accW[(size_t)n * HD + d]      = w0;
  accW[(size_t)n * HD + 32 + d] = w1;
  accW[(size_t)n * HD + 64 + d] = w2;
}

// ---- finish: h_next = relu((accW @ Wstack)/H + meanV + x@Ws+bs + res)
//      Wstack[h*32+i][j] = We[i][h*32+j]
__global__ __launch_bounds__(32)
void k_finish(const float* __restrict__ accW, const float* __restrict__ We,
              const float* __restrict__ meanV, const float* __restrict__ sbuf,
              const float* __restrict__ res, float* __restrict__ hout) {
  __shared__ float As[16][HD];
  const int lane = threadIdx.x;
  const int r0 = blockIdx.x * 16;
  for (int idx = lane; idx < 16 * HD; idx += 32) {
    int r = idx / HD, f = idx % HD;
    int row = r0 + r;
    As[r][f] = (row < NN) ? accW[(size_t)row * HD + f] : 0.f;
  }
  __syncthreads();
  const int lh = lane >> 4, lm = lane & 15;
  for (int ct = 0; ct < 2; ++ct) {
    v8f c = {};
    for (int k0 = 0; k0 < HD; k0 += 4) {
      int kr = k0 + 2 * lh;
      int i0 = kr & 31, h0 = kr >> 5;
      int i1 = (kr + 1) & 31, h1 = (kr + 1) >> 5;
      v2f a; a.x = As[lm][kr]; a.y = As[lm][kr + 1];
      v2f b; b.x = We[i0 * HD + h0 * 32 + ct * 16 + lm];
      b.y = We[i1 * HD + h1 * 32 + ct * 16 + lm];
      c = wmma_f32(a, b, c);
    }
    for (int r = 0; r < 8; ++r) {
      int row = r0 + 8 * lh + r;
      if (row < NN) {
        int col = ct * 16 + lm;
        size_t o = (size_t)row * DOUT + col;
        float y = c[r] * (1.f / 3.f) + meanV[o] + sbuf[o] + res[o];
        hout[o] = fmaxf(y, 0.f);
      }
    }
  }
}

// ---------------- final head: out = concat([h, h_cur]) @ ctr_W + ctr_b
__global__ __launch_bounds__(32)
void k_final(const float* __restrict__ h, const float* __restrict__ hc,
             const float* __restrict__ W, const float* __restrict__ bias,
             float* __restrict__ out) {
  __shared__ float As[16][HD];
  const int lane = threadIdx.x;
  const int r0 = blockIdx.x * 16;
  for (int idx = lane; idx < 16 * HD; idx += 32) {
    int r = idx / HD, f = idx % HD;
    int row = r0 + r;
    float val = 0.f;
    if (row < NN) val = (f < DIN) ? h[(size_t)row * DIN + f]
                                  : hc[(size_t)row * DOUT + (f - DIN)];
    As[r][f] = val;
  }
  __syncthreads();
  const int lh = lane >> 4, lm = lane & 15;
  for (int ct = 0; ct < 2; ++ct) {
    v8f c = {};
    for (int k0 = 0; k0 < HD; k0 += 4) {
      int kr = k0 + 2 * lh;
      v2f a; a.x = As[lm][kr]; a.y = As[lm][kr + 1];
      v2f b; b.x = W[kr * DOUT + ct * 16 + lm];
      b.y = W[(kr + 1) * DOUT + ct * 16 + lm];
      c = wmma_f32(a, b, c);
    }
    for (int r = 0; r < 8; ++r) {
      int row = r0 + 8 * lh + r;
      if (row < NN) out[(size_t)row * DOUT + ct * 16 + lm] = c[r] + bias[ct * 16 + lm];
    }
  }
}

extern "C" void kernel_launch(void* const* d_in, const int* in_sizes, int n_in,
                              void* d_out, int out_size, void* d_ws, size_t ws_size,
                              hipStream_t stream) {
  const float* h    = (const float*)d_in[0];
  const float* geo  = (const float*)d_in[1];
  const float* adj  = (const float*)d_in[2];
  const float* eW   = (const float*)d_in[3];
  const float* eb   = (const float*)d_in[4];
  const float* resW = (const float*)d_in[5];
  const float* resb = (const float*)d_in[6];
  const float* ctrW = (const float*)d_in[34];
  const float* ctrb = (const float*)d_in[35];
  float* out = (float*)d_out;

  size_t off = 0;
  auto alloc = [&](size_t bytes) {
    void* p = (char*)d_ws + off;
    off += (bytes + 255) & ~(size_t)255;
    return p;
  };
  int*   src_i = (int*)alloc((size_t)EE * 4);
  int*   dst_i = (int*)alloc((size_t)EE * 4);
  int*   rs    = (int*)alloc((size_t)(NN + 1) * 4);
  float* dw    = (float*)alloc((size_t)EE * 4);
  float* padW  = (float*)alloc((size_t)EFKP * DOUT * 4);
  float* eattr = (float*)alloc((size_t)EE * DOUT * 4);
  float* alpha = (float*)alloc((size_t)EE * HH * 4);
  float* qb    = (float*)alloc((size_t)NN * HD * 4);
  float* kb    = (float*)alloc((size_t)NN * HD * 4);
  float* vb    = (float*)alloc((size_t)NN * HD * 4);
  float* sbuf  = (float*)alloc((size_t)NN * DOUT * 4);
  float* resb_buf = (float*)alloc((size_t)NN * DOUT * 4);
  float* amax  = (float*)alloc((size_t)NN * HH * 4);
  float* den   = (float*)alloc((size_t)NN * HH * 4);
  float* meanV = (float*)alloc((size_t)NN * DOUT * 4);
  float* accW  = (float*)alloc((size_t)NN * HD * 4);
  float* hb0   = (float*)alloc((size_t)NN * DOUT * 4);
  float* hb1   = (float*)alloc((size_t)NN * DOUT * 4);

  const int ET = (EE + 15) / 16;   // edge tiles
  const int NT = (NN + 15) / 16;   // node tiles

  k_prep<<<(EE + 255) / 256, 256, 0, stream>>>(adj, src_i, dst_i, dw);
  k_padW<<<(EFKP * DOUT + 255) / 256, 256, 0, stream>>>(eW, padW);
  k_rowstart<<<(NN + 256) / 256, 256, 0, stream>>>(dst_i, rs);
  k_edge_attr<<<ET, 32, 0, stream>>>(h, geo, src_i, dst_i, dw, padW, eb, eattr);

  for (int i = 0; i < 3; ++i) {
    const int base = 7 + i * 9;
    const float* Wq = (const float*)d_in[base + 0];
    const float* bq = (const float*)d_in[base + 1];
    const float* Wk = (const float*)d_in[base + 2];
    const float* bk = (const float*)d_in[base + 3];
    const float* Wv = (const float*)d_in[base + 4];
    const float* bv = (const float*)d_in[base + 5];
    const float* We = (const float*)d_in[base + 6];
    const float* Ws = (const float*)d_in[base + 7];
    const float* bs = (const float*)d_in[base + 8];

    const int fin = (i == 0) ? DIN : DOUT;
    const float* x = (i == 0) ? h : ((i == 1) ? hb0 : hb1);
    const float* res = (i == 0) ? resb_buf : ((i == 1) ? hb0 : hb1);
    float* hout = (i == 1) ? hb1 : hb0;  // layer0->hb0, layer1->hb1, layer2->hb0
    const size_t shA = (size_t)16 * fin * 4;

    k_gemm_node<<<NT, 32, shA, stream>>>(x, Wq, bq, qb, NN, fin, HD);
    k_gemm_node<<<NT, 32, shA, stream>>>(x, Wk, bk, kb, NN, fin, HD);
    k_gemm_node<<<NT, 32, shA, stream>>>(x, Wv, bv, vb, NN, fin, HD);
    k_gemm_node<<<NT, 32, shA, stream>>>(x, Ws, bs, sbuf, NN, fin, DOUT);
    if (i == 0)
      k_gemm_node<<<NT, 32, (size_t)16 * DIN * 4, stream>>>(
          h, resW, resb, resb_buf, NN, DIN, DOUT);

    k_alpha<<<ET, 32, 0, stream>>>(eattr, We, qb, kb, src_i, dst_i, alpha);
    k_stats<<<(NN * HH + 255) / 256, 256, 0, stream>>>(alpha, rs, amax, den);
    k_norm<<<(EE * HH + 255) / 256, 256, 0, stream>>>(alpha, dst_i, amax, den);
    k_aggregate<<<(NN * DOUT + 255) / 256, 256, 0, stream>>>(alpha, rs, src_i, vb,
                                                             eattr, meanV, accW);
    k_finish<<<NT, 32, 0, stream>>>(accW, We, meanV, sbuf, res, hout);
  }

  // final head uses original h and layer-2 output (hb0)
  k_final<<<NT, 32, 0, stream>>>(h, hb0, ctrW, ctrb, out);
}